// ConvexHullGLGMLP_76948634075339
// MI455X (gfx1250) — compile-verified
//
#include <hip/hip_runtime.h>
#include <math.h>

// ---------------------------------------------------------------------------
// Geometric-algebra MLP forward, fully fused. One workgroup per batch element.
// f32 WMMA (V_WMMA_F32_16X16X4_F32) for the per-layer channel-mixing GEMMs.
// State + Cayley/weight tables resident in LDS; no HBM traffic between layers.
// ---------------------------------------------------------------------------

#define GA_DIM   64
#define GA_B     256
#define GA_INF   16
#define GA_HID   64
#define GA_NGP   4
#define GA_STR   66   // padded LDS row stride (floats): bank = (2*row + col) % 64
#define GA_WST   65   // padded gp_w row stride: bank = (row + col) % 64

typedef __attribute__((ext_vector_type(2))) float v2f;
typedef __attribute__((ext_vector_type(8))) float v8f;

// ----------------------------- compile-time algebra -------------------------
constexpr int cpopc(int x) { int c = 0; while (x) { c += x & 1; x >>= 1; } return c; }

struct Alg {
    short masks[64];
    short pos[64];
    short qt[64];
    unsigned short kqs[64][64];  // [j][i]: k(6b) | qidx(6b)<<6 | sign(1b)<<12
};

constexpr Alg buildAlg() {
    Alg a{};
    int idx = 0;
    for (int pc = 0; pc <= 6; ++pc)
        for (int m = 0; m < 64; ++m)
            if (cpopc(m) == pc) { a.masks[idx] = (short)m; ++idx; }
    for (int i = 0; i < 64; ++i) a.pos[a.masks[i]] = (short)i;
    for (int i = 0; i < 64; ++i) a.qt[i] = (short)(cpopc(a.masks[i]) & 3);
    for (int j = 0; j < 64; ++j) {
        for (int i = 0; i < 64; ++i) {
            const int am = a.masks[i];
            const int jm = a.masks[j];
            const int bm = am ^ jm;          // mask of k with pos[am^bm] == j
            const int k  = a.pos[bm];
            int s = 0, t = am >> 1;
            while (t) { s += cpopc(t & bm); t >>= 1; }
            const int qidx = a.qt[i] * 16 + a.qt[j] * 4 + a.qt[k];
            a.kqs[j][i] = (unsigned short)(k | (qidx << 6) | ((s & 1) << 12));
        }
    }
    return a;
}
constexpr Alg ALG = buildAlg();

constexpr int qcnt(int q) { int c = 0; for (int i = 0; i < 64; ++i) if (ALG.qt[i] == q) ++c; return c; }
static_assert(qcnt(0) == 16 && qcnt(1) == 12 && qcnt(2) == 16 && qcnt(3) == 20, "qt group sizes");

// Components are popcount-sorted, so each qt-group tile row maps to a component
// via pure arithmetic (pad rows -> dummy LDS row 64). mt is wave-uniform.
//   tile0: qt0 = {0} u {42..56}      tile1: qt1 = {1..6} u {57..62} (+4 pad)
//   tile2: qt2 = {7..21} u {63}      tile3: qt3[0:16] = {22..37}
//   tile4: qt3[16:20] = {38..41} (+12 pad)
constexpr int tileRow(int mt, int m) {
    if (mt == 0) return (m == 0) ? 0 : 41 + m;
    if (mt == 1) return (m < 12) ? ((m < 6) ? 1 + m : 51 + m) : 64;
    if (mt == 2) return (m < 15) ? 7 + m : 63;
    if (mt == 3) return 22 + m;
    return (m < 4) ? 38 + m : 64;
}

constexpr bool checkTiles() {
    int lists[4][20] = {};
    int cnt[4] = {0, 0, 0, 0};
    for (int i = 0; i < 64; ++i) { const int q = ALG.qt[i]; lists[q][cnt[q]] = i; ++cnt[q]; }
    const int order[5] = {0, 1, 2, 3, 3};
    const int start[5] = {0, 0, 0, 0, 16};
    for (int mt = 0; mt < 5; ++mt)
        for (int m = 0; m < 16; ++m) {
            const int src = start[mt] + m;
            const int expect = (src < cnt[order[mt]]) ? lists[order[mt]][src] : 64;
            if (tileRow(mt, m) != expect) return false;
        }
    return true;
}
static_assert(checkTiles(), "tile row formula mismatch");

// ----------------------------- kernels --------------------------------------
__global__ __launch_bounds__(256) void ga_forward(
    const float* __restrict__ points,    // (B,16,6)
    const float* __restrict__ products,  // (B,)
    const float* __restrict__ lin_w,     // (64,16,4)
    const float* __restrict__ lin_b,     // (64,)
    const float* __restrict__ gp_lin_w,  // (4,64,64,4)
    const float* __restrict__ gp_a,      // (4,64,4)
    const float* __restrict__ gp_w,     // (4,64,4,4,4)
    const float* __restrict__ mlp_w1,    // (64,64)
    const float* __restrict__ mlp_b1,    // (64,)
    const float* __restrict__ mlp_w2,    // (1,64)
    const float* __restrict__ mlp_b2,    // (1,)
    float* __restrict__ out)             // [mean, loss(256)]
{
    __shared__ float Xc[(GA_DIM + 1) * GA_STR];        // state (row 64 = zero dummy)
    __shared__ float XR[(GA_DIM + 1) * GA_STR];        // xr    (row 64 = pad sink)
    __shared__ float Wl[GA_HID * GA_WST];              // gp_w layer slice, padded
    __shared__ unsigned short KQ[64 * 64];             // kqs table [j*64+i]
    __shared__ float ys[GA_HID];
    __shared__ float hs[GA_HID];

    const int b    = blockIdx.x;
    const int t    = threadIdx.x;
    const int wave = t >> 5;
    const int lane = t & 31;
    const int m16  = lane & 15;   // A-row / B-col / D-col within a 16-tile
    const int kh   = lane >> 4;   // K half (0: K=0,1 ; 1: K=2,3), D row half

    // zero state buffer (incl. dummy row), stage Cayley table into LDS
    for (int idx = t; idx < (GA_DIM + 1) * GA_STR; idx += 256) Xc[idx] = 0.0f;
    for (int idx = t; idx < 64 * 64; idx += 256)
        KQ[idx] = (unsigned short)ALG.kqs[idx >> 6][idx & 63];
    __syncthreads();

    // --- embedding + first qt_linear (scalar-comp bias + 6 grade-1 comps)
    if (t < GA_HID) {
        const int f = t;
        Xc[0 * GA_STR + f] = lin_b[f];
        for (int d = 0; d < 6; ++d) {
            float acc = 0.0f;
            for (int m = 0; m < GA_INF; ++m)
                acc = fmaf(points[(b * GA_INF + m) * 6 + d],
                           lin_w[(f * GA_INF + m) * 4 + 1], acc);  // qt(grade-1)=1
            Xc[(1 + d) * GA_STR + f] = acc;
        }
    }
    __syncthreads();

    for (int l = 0; l < GA_NGP; ++l) {
        // ---- stage gp_w layer slice into padded LDS (used by Phase C)
        {
            const float* gw = gp_w + (size_t)l * GA_HID * 64;
            for (int idx = t; idx < GA_HID * 64; idx += 256)
                Wl[(idx >> 6) * GA_WST + (idx & 63)] = gw[idx];
        }

        // ---- Phase A: xr_pre[i,f'] = sum_f Xc[i,f] * lw[f',f,qt[i]] via f32 WMMA
        for (int job = wave; job < 20; job += 8) {          // wave-uniform loop
            const int mt = job >> 2;                        // 0..4 (M tile)
            const int nt = job & 3;                         // 0..3 (N tile)
            const int c  = (mt == 4) ? 3 : mt;              // qt of this tile
            const int rowA = tileRow(mt, m16);              // comp (or dummy 64)
            const int n    = nt * 16 + m16;                 // output channel
            const float* arow = Xc + rowA * GA_STR;
            const float* wb   = gp_lin_w + ((size_t)(l * GA_HID + n) * GA_HID) * 4 + c;

            v8f acc = {0.f, 0.f, 0.f, 0.f, 0.f, 0.f, 0.f, 0.f};
            for (int k0 = 0; k0 < GA_HID; k0 += 4) {
                const int k = k0 + 2 * kh;
                v2f av; av.x = arow[k];           av.y = arow[k + 1];
                v2f bv; bv.x = wb[(size_t)k * 4]; bv.y = wb[(size_t)(k + 1) * 4];
                acc = __builtin_amdgcn_wmma_f32_16x16x4_f32(
                          false, av, false, bv, (short)0, acc, false, false);
            }
#pragma unroll
            for (int r = 0; r < 8; ++r) {                   // branchless D store
                const int m = r + 8 * kh;
                XR[tileRow(mt, m) * GA_STR + nt * 16 + m16] = acc[r];
            }
        }
        __syncthreads();

        // ---- Phase B: qt_normalize(xr_pre, gp_a[l]); qt is a step fn of comp idx
        if (t < GA_HID) {
            const int f = t;
            float s0 = 0.f, s1 = 0.f, s2 = 0.f, s3 = 0.f;
            {
                float v;
                v = XR[0 * GA_STR + f]; s0 = fmaf(v, v, s0);
                for (int i = 1;  i < 7;  ++i) { v = XR[i * GA_STR + f]; s1 = fmaf(v, v, s1); }
                for (int i = 7;  i < 22; ++i) { v = XR[i * GA_STR + f]; s2 = fmaf(v, v, s2); }
                for (int i = 22; i < 42; ++i) { v = XR[i * GA_STR + f]; s3 = fmaf(v, v, s3); }
                for (int i = 42; i < 57; ++i) { v = XR[i * GA_STR + f]; s0 = fmaf(v, v, s0); }
                for (int i = 57; i < 63; ++i) { v = XR[i * GA_STR + f]; s1 = fmaf(v, v, s1); }
                v = XR[63 * GA_STR + f]; s2 = fmaf(v, v, s2);
            }
            float sc[4];
            const float ssum[4] = {s0, s1, s2, s3};
#pragma unroll
            for (int c = 0; c < 4; ++c) {
                float nm = sqrtf(fmaxf(ssum[c], 1e-12f));
                const float av = gp_a[(l * GA_HID + f) * 4 + c];
                const float sg = 1.0f / (1.0f + __expf(-av));
                nm = sg * (nm - 1.0f) + 1.0f;
                sc[c] = nm + 1e-6f;
            }
            XR[0 * GA_STR + f] /= sc[0];
            for (int i = 1;  i < 7;  ++i) XR[i * GA_STR + f] /= sc[1];
            for (int i = 7;  i < 22; ++i) XR[i * GA_STR + f] /= sc[2];
            for (int i = 22; i < 42; ++i) XR[i * GA_STR + f] /= sc[3];
            for (int i = 42; i < 57; ++i) XR[i * GA_STR + f] /= sc[0];
            for (int i = 57; i < 63; ++i) XR[i * GA_STR + f] /= sc[1];
            XR[63 * GA_STR + f] /= sc[2];
        }
        __syncthreads();

        // ---- Phase C: geometric product, out[j,f] = sum_i sign*w*x[i,f]*xr[k,f]
        {
            const int f  = t & 63;
            const int jg = t >> 6;                          // 0..3, wave-uniform
            float acc[16];
#pragma unroll
            for (int jj = 0; jj < 16; ++jj) acc[jj] = 0.0f;
            for (int i = 0; i < GA_DIM; ++i) {
                const float xv = Xc[i * GA_STR + f];
#pragma unroll
                for (int jj = 0; jj < 16; ++jj) {
                    const unsigned e = KQ[(jg * 16 + jj) * 64 + i];
                    const float wv = Wl[f * GA_WST + ((e >> 6) & 63)];
                    const float p  = xv * XR[(e & 63) * GA_STR + f] * wv;
                    acc[jj] += (e & 4096u) ? -p : p;
                }
            }
            __syncthreads();                                // all reads of Xc done
#pragma unroll
            for (int jj = 0; jj < 16; ++jj)
                Xc[(jg * 16 + jj) * GA_STR + f] = acc[jj];
        }
        __syncthreads();
    }

    // ---- component norm -> MLP head
    if (t < GA_HID) {
        float s = 0.0f;
        for (int i = 0; i < GA_DIM; ++i) {
            const float v = Xc[i * GA_STR + t];
            s = fmaf(v, v, s);
        }
        ys[t] = sqrtf(s);
    }
    __syncthreads();
    if (t < GA_HID) {
        float acc = mlp_b1[t];
        for (int f = 0; f < GA_HID; ++f)
            acc = fmaf(ys[f], mlp_w1[t * GA_HID + f], acc);
        const float h = acc / (1.0f + __expf(-acc));        // silu
        hs[t] = h * mlp_w2[t];
    }
    __syncthreads();
    if (t == 0) {
        float pred = mlp_b2[0];
        for (int n = 0; n < GA_HID; ++n) pred += hs[n];
        const float d = pred - products[b];
        out[1 + b] = d * d;
    }
}

__global__ void ga_mean(float* __restrict__ out) {
    if (threadIdx.x == 0) {
        float s = 0.0f;
        for (int i = 0; i < GA_B; ++i) s += out[1 + i];
        out[0] = s * (1.0f / (float)GA_B);
    }
}

// ----------------------------- launcher -------------------------------------
extern "C" void kernel_launch(void* const* d_in, const int* in_sizes, int n_in,
                              void* d_out, int out_size, void* d_ws, size_t ws_size,
                              hipStream_t stream) {
    const float* points   = (const float*)d_in[0];
    const float* products = (const float*)d_in[1];
    const float* lin_w    = (const float*)d_in[2];
    const float* lin_b    = (const float*)d_in[3];
    const float* gp_lin_w = (const float*)d_in[4];
    const float* gp_a     = (const float*)d_in[5];
    const float* gp_w     = (const float*)d_in[6];
    const float* mlp_w1   = (const float*)d_in[7];
    const float* mlp_b1   = (const float*)d_in[8];
    const float* mlp_w2   = (const float*)d_in[9];
    const float* mlp_b2   = (const float*)d_in[10];
    float* out = (float*)d_out;

    ga_forward<<<GA_B, 256, 0, stream>>>(points, products, lin_w, lin_b,
                                         gp_lin_w, gp_a, gp_w,
                                         mlp_w1, mlp_b1, mlp_w2, mlp_b2, out);
    ga_mean<<<1, 32, 0, stream>>>(out);
}